// RNNTLoss_65008624992373
// MI455X (gfx1250) — compile-verified
//
#include <hip/hip_runtime.h>
#include <math.h>

#define BB 16
#define TT 128
#define UU 32
#define VV 1024
#define UP1 (UU + 1)

#if defined(__has_builtin)
#if __has_builtin(__builtin_amdgcn_global_load_async_to_lds_b128)
#define RNNT_ASYNC 1
#endif
#endif

typedef int v4i __attribute__((vector_size(16)));
typedef __attribute__((address_space(1))) v4i* as1_v4i_p;
typedef __attribute__((address_space(3))) v4i* as3_v4i_p;

__device__ __forceinline__ float warp_max_f(float v) {
#pragma unroll
    for (int o = 16; o > 0; o >>= 1) v = fmaxf(v, __shfl_xor(v, o, 32));
    return v;
}
__device__ __forceinline__ float warp_sum_f(float v) {
#pragma unroll
    for (int o = 16; o > 0; o >>= 1) v += __shfl_xor(v, o, 32);
    return v;
}
__device__ __forceinline__ float log_add_exp(float a, float b) {
    float m = fmaxf(a, b);
    if (m == -__builtin_inff()) return m;
    return m + log1pf(__expf(-fabsf(a - b)));
}

// ---------------- kernel 0: exclusive cumsum of label_lens ----------------
__global__ void rnnt_prep_kernel(const int* __restrict__ label_lens, int* __restrict__ cum) {
    if (threadIdx.x == 0 && blockIdx.x == 0) {
        int acc = 0;
        for (int b = 0; b < BB; ++b) { cum[b] = acc; acc += label_lens[b]; }
    }
}

// ---------------- kernel 1: streaming logsumexp + gather (memory-bound) ----------------
// One wave32 per row of V=1024 floats (4KB). 8 x b128 loads per lane.
__global__ __launch_bounds__(256) void rnnt_lse_kernel(
    const float* __restrict__ acts, const int* __restrict__ labels,
    const int* __restrict__ blank_p, const int* __restrict__ cum,
    float* __restrict__ blank_lp, float* __restrict__ emit_lp) {
    const int wave = threadIdx.x >> 5;
    const int lane = threadIdx.x & 31;
    const int row  = blockIdx.x * 8 + wave;   // R = B*T*(U+1) = 67584, divisible by 8
    const size_t base = (size_t)row * VV;
    const float4* p4 = (const float4*)(acts + base);

    float4 v[8];
#pragma unroll
    for (int k = 0; k < 8; ++k) v[k] = p4[k * 32 + lane];

    float m = -__builtin_inff();
#pragma unroll
    for (int k = 0; k < 8; ++k)
        m = fmaxf(m, fmaxf(fmaxf(v[k].x, v[k].y), fmaxf(v[k].z, v[k].w)));
    m = warp_max_f(m);

    float s = 0.f;
#pragma unroll
    for (int k = 0; k < 8; ++k) {
        s += __expf(v[k].x - m);
        s += __expf(v[k].y - m);
        s += __expf(v[k].z - m);
        s += __expf(v[k].w - m);
    }
    s = warp_sum_f(s);
    const float lse = m + __logf(s);

    if (lane == 0) {
        const int u  = row % UP1;
        const int bt = row / UP1;     // b*T + t
        const int b  = bt / TT;
        const int blank = *blank_p;
        blank_lp[row] = acts[base + blank] - lse;
        if (u < UU) {
            int idx = cum[b] + u;
            idx = min(max(idx, 0), BB * UU - 1);
            const int lab = labels[idx];
            emit_lp[bt * UU + u] = acts[base + lab] - lse;
        }
    }
}

// ---------------- kernel 2: wavefront DP per batch ----------------
__global__ __launch_bounds__(64) void rnnt_dp_kernel(
    const float* __restrict__ blank_lp, const float* __restrict__ emit_lp,
    const int* __restrict__ act_lens, const int* __restrict__ label_lens,
    float* __restrict__ costs) {
    __shared__ __align__(16) float sblank[TT * UP1];  // 4224 floats
    __shared__ __align__(16) float semit[TT * UU];    // 4096 floats
    __shared__ float diag[UP1];
    __shared__ float s_ll;

    const int b   = blockIdx.x;
    const int tid = threadIdx.x;
    const float* gblank = blank_lp + (size_t)b * TT * UP1;
    const float* gemit  = emit_lp  + (size_t)b * TT * UU;

#ifdef RNNT_ASYNC
    // CDNA5 async copy: global -> LDS (tracked by ASYNCcnt, no VGPR data movement)
    for (int i = tid; i < (TT * UP1) / 4; i += 64)
        __builtin_amdgcn_global_load_async_to_lds_b128(
            (as1_v4i_p)(void*)(gblank + 4 * i),
            (as3_v4i_p)(void*)(sblank + 4 * i), 0, 0);
    for (int i = tid; i < (TT * UU) / 4; i += 64)
        __builtin_amdgcn_global_load_async_to_lds_b128(
            (as1_v4i_p)(void*)(gemit + 4 * i),
            (as3_v4i_p)(void*)(semit + 4 * i), 0, 0);
#if defined(__has_builtin) && __has_builtin(__builtin_amdgcn_s_wait_asynccnt)
    __builtin_amdgcn_s_wait_asynccnt(0);
#else
    asm volatile("s_wait_asynccnt 0" ::: "memory");
#endif
#else
    for (int i = tid; i < (TT * UP1) / 4; i += 64)
        ((float4*)sblank)[i] = ((const float4*)gblank)[i];
    for (int i = tid; i < (TT * UU) / 4; i += 64)
        ((float4*)semit)[i] = ((const float4*)gemit)[i];
#endif

    const int Tb = act_lens[b];
    const int Ub = label_lens[b];
    const int u  = tid;
    if (u <= UU) diag[u] = (u == 0) ? 0.f : -__builtin_inff();
    float cur = (u == 0) ? 0.f : -__builtin_inff();
    __syncthreads();

    // Anti-diagonal wavefront: lane u computes alpha[d-u][u] at diagonal d.
    // Own register 'cur' = alpha[t-1][u]; diag[u-1] = alpha[t][u-1].
    for (int d = 1; d <= (TT - 1) + UU; ++d) {
        const int t = d - u;
        float left = (u >= 1 && u <= UU) ? diag[u - 1] : -__builtin_inff();
        __syncthreads();
        if (u <= UU && t >= 0 && t < TT) {
            float val;
            if (u == 0) {
                val = cur + sblank[(t - 1) * UP1];            // t >= 1 here
            } else if (t == 0) {
                val = left + semit[u - 1];                    // emit[0][u-1]
            } else {
                val = log_add_exp(cur + sblank[(t - 1) * UP1 + u],
                                  left + semit[t * UU + (u - 1)]);
            }
            cur = val;
            diag[u] = val;
            if (t == Tb - 1 && u == Ub) s_ll = val + sblank[t * UP1 + u];
        }
        __syncthreads();
    }
    if (tid == 0) costs[b] = -s_ll;
}

// ---------------- kernel 3: final sum ----------------
__global__ void rnnt_sum_kernel(const float* __restrict__ costs, float* __restrict__ out) {
    const int lane = threadIdx.x;
    float v = (lane < BB) ? costs[lane] : 0.f;
    v = warp_sum_f(v);
    if (lane == 0) out[0] = v;
}

extern "C" void kernel_launch(void* const* d_in, const int* in_sizes, int n_in,
                              void* d_out, int out_size, void* d_ws, size_t ws_size,
                              hipStream_t stream) {
    (void)in_sizes; (void)n_in; (void)out_size; (void)ws_size;
    const float* acts     = (const float*)d_in[0];
    const int* labels     = (const int*)d_in[1];
    const int* act_lens   = (const int*)d_in[2];
    const int* label_lens = (const int*)d_in[3];
    const int* blank      = (const int*)d_in[4];

    float* ws       = (float*)d_ws;
    float* blank_lp = ws;                           // B*T*(U+1) = 67584 floats
    float* emit_lp  = blank_lp + BB * TT * UP1;     // B*T*U     = 65536 floats
    float* costs    = emit_lp + BB * TT * UU;       // B         = 16 floats
    int*   cum      = (int*)(costs + BB);           // B ints

    rnnt_prep_kernel<<<1, 32, 0, stream>>>(label_lens, cum);

    const int rows = BB * TT * UP1;                 // 67584
    rnnt_lse_kernel<<<rows / 8, 256, 0, stream>>>(acts, labels, blank, cum,
                                                  blank_lp, emit_lp);

    rnnt_dp_kernel<<<BB, 64, 0, stream>>>(blank_lp, emit_lp, act_lens, label_lens, costs);

    rnnt_sum_kernel<<<1, 32, 0, stream>>>(costs, (float*)d_out);
}